// ApproxCompressor_73564199846489
// MI455X (gfx1250) — compile-verified
//
#include <hip/hip_runtime.h>

// ---------------------------------------------------------------------------
// ApproxCompressor on MI455X (gfx1250, wave32).
//
// envelope = energy (*) (1-a)*a^n  (K=16384 taps) computed as a first-order
// IIR scan  y[t] = a*y[t-1] + (1-a)*energy[t]  with exact truncation fix
// envelope[t] = y[t] - a^K * y[t-K].  Intra-chunk (16-sample) work is a 16x16
// lower-triangular Toeplitz matmul done with V_WMMA_F32_16X16X4_F32 (fp32).
// Memory-bound: ~135 MB HBM traffic @ 23.3 TB/s ~ 6 us.
// ---------------------------------------------------------------------------

typedef float v2f __attribute__((ext_vector_type(2)));
typedef float v4f __attribute__((ext_vector_type(4)));
typedef float v8f __attribute__((ext_vector_type(8)));

#define N_BATCH 32
#define L_LEN   131072
#define K_IIR   16384
#define SEG     4096
#define NSEG    (L_LEN / SEG)     // 32 segments per row
#define NCHUNK  (SEG / 16)        // 256 chunks per segment
#define P_STRIDE 12
#define EPSV    1e-5f

// workspace layout (float offsets)
#define WS_Y      ((size_t)0)                              // N*L local-scan y
#define WS_SEGEND ((size_t)N_BATCH * L_LEN)                // N*NSEG
#define WS_CARRY  (WS_SEGEND + (size_t)N_BATCH * NSEG)     // N*NSEG
#define WS_PAR    (WS_CARRY + (size_t)N_BATCH * NSEG)      // N*P_STRIDE

// ---------------------------------------------------------------------------
__global__ void k_params(const float* __restrict__ z_alpha,
                         const float* __restrict__ log_thr,
                         const float* __restrict__ log_ratio,
                         const float* __restrict__ log_knee,
                         float* __restrict__ ws) {
  int b = threadIdx.x;
  if (b >= N_BATCH) return;
  float za = z_alpha[b];
  float alpha = 1.0f / (1.0f + expf(-za));
  alpha = fminf(fmaxf(alpha, 1e-12f), 1.0f - 1e-7f);
  float la = logf(alpha);
  float* p = ws + WS_PAR + (size_t)b * P_STRIDE;
  p[0] = alpha;                       // a
  p[1] = 1.0f - alpha;                // (1-a)
  p[2] = expf(16.0f * la);            // a^16   (chunk decay)
  p[3] = expf((float)SEG * la);       // a^4096 (segment decay)
  p[4] = expf((float)K_IIR * la);     // a^K    (truncation weight)
  p[5] = la;
  p[6] = log_thr[b] - 6.0f;           // T
  p[7] = 1.0f / (1.0f + expf(log_ratio[b])) - 1.0f;  // c = 1/R - 1
  p[8] = expf(log_knee[b]);           // W
}

// ---------------------------------------------------------------------------
// Per-(row, segment) local scan.  256 threads = 8 waves.  Each wave runs the
// 16x16 Toeplitz conv on two 256-sample units via 4 accumulating fp32 WMMAs;
// chunk carries combined with a uniform-decay Hillis-Steele scan in LDS.
__global__ __launch_bounds__(256) void k_localscan(const float* __restrict__ x,
                                                   float* __restrict__ ws) {
  __shared__ float ls_e[SEG];       // energy
  __shared__ float ls_y[SEG];       // local y
  __shared__ float ls_s[NCHUNK];    // chunk-end scan
  __shared__ float ls_tri[32];      // [16+d] = (1-a)*a^d (d=0..15), [<16] = 0
  __shared__ float ls_ap[17];       // a^0 .. a^16

  const int g = blockIdx.x;         // segment
  const int r = blockIdx.y;         // batch row
  const int tid = threadIdx.x;
  const int lane = tid & 31;
  const int wv = tid >> 5;

  const float* p = ws + WS_PAR + (size_t)r * P_STRIDE;
  const float one_m = p[1];
  const float a16 = p[2];
  const float la = p[5];

  if (tid < 16) {
    float ad = expf((float)tid * la);    // a^tid
    ls_tri[tid] = 0.0f;
    ls_tri[16 + tid] = one_m * ad;
    ls_ap[tid] = ad;
  }
  if (tid == 16) ls_ap[16] = expf(16.0f * la);

  const size_t segbase = (size_t)g * SEG;
  const float* x0 = x + (size_t)r * 2 * L_LEN;
  const float* x1 = x0 + L_LEN;

  // coalesced energy load: energy[t] = (x0^2 + x1^2)/2
  for (int i = 0; i < 16; ++i) {
    size_t t = segbase + (size_t)(i * 256 + tid);
    float a = x0[t];
    float b2 = x1[t];
    ls_e[i * 256 + tid] = 0.5f * (a * a + b2 * b2);
  }
  __syncthreads();

  // ---- WMMA: D = T(16x16 lower-tri Toeplitz) * X(16 pos x 16 chunks) ----
  const int half = lane >> 4;       // 0: lanes 0-15, 1: lanes 16-31
  const int col = lane & 15;        // A-row (M) / B-col (N) / D-col (N)

  // Hoist A operands (loop-invariant per wave): branch-free via padded table.
  // A[M=col][K=ka+q] = ls_tri[16 + col - ka - q]
  v2f Areg[4];
#pragma unroll
  for (int kb = 0; kb < 4; ++kb) {
    int ka = kb * 4 + half * 2;
    Areg[kb].x = ls_tri[16 + col - ka];
    Areg[kb].y = ls_tri[15 + col - ka];
  }

  for (int j = 0; j < 2; ++j) {
    int ub = (wv * 2 + j) * 256;            // 16 units of 256 samples
    // B pair base for this lane: chunk 'col', position 2*half (even -> b64 ok)
    const v2f* bp = (const v2f*)&ls_e[ub + col * 16 + half * 2];
    v2f B0 = bp[0];                          // K = 0*4 + 2h, +1
    v2f B1 = bp[2];                          // K = 1*4 + 2h, +1
    v2f B2 = bp[4];                          // K = 2*4 + 2h, +1
    v2f B3 = bp[6];                          // K = 3*4 + 2h, +1

    v8f acc = {0.f, 0.f, 0.f, 0.f, 0.f, 0.f, 0.f, 0.f};
    acc = __builtin_amdgcn_wmma_f32_16x16x4_f32(false, Areg[0], false, B0,
                                                (short)0, acc, false, false);
    acc = __builtin_amdgcn_wmma_f32_16x16x4_f32(false, Areg[1], false, B1,
                                                (short)0, acc, false, false);
    acc = __builtin_amdgcn_wmma_f32_16x16x4_f32(false, Areg[2], false, B2,
                                                (short)0, acc, false, false);
    acc = __builtin_amdgcn_wmma_f32_16x16x4_f32(false, Areg[3], false, B3,
                                                (short)0, acc, false, false);

    // D rows for this lane are 8 consecutive samples -> two b128 stores
    int db = ub + col * 16 + 8 * half;       // 32-byte aligned
    v4f lo = {acc[0], acc[1], acc[2], acc[3]};
    v4f hi = {acc[4], acc[5], acc[6], acc[7]};
    *(v4f*)&ls_y[db] = lo;
    *(v4f*)&ls_y[db + 4] = hi;
  }
  __syncthreads();

  // ---- Hillis-Steele scan of chunk-end states: s_c = e_c + a^16 * s_{c-1} --
  float s = ls_y[tid * 16 + 15];
  ls_s[tid] = s;
  __syncthreads();
  float mp = a16;                   // a^(16*step)
  for (int st = 1; st < NCHUNK; st <<= 1) {
    float add = (tid >= st) ? ls_s[tid - st] : 0.0f;
    __syncthreads();
    s += mp * add;
    ls_s[tid] = s;
    mp *= mp;
    __syncthreads();
  }

  // apply carry within own chunk: y[i] += a^(i+1) * s_{c-1}
  float carry = (tid > 0) ? ls_s[tid - 1] : 0.0f;
#pragma unroll
  for (int i = 0; i < 16; ++i) {
    ls_y[tid * 16 + i] += ls_ap[i + 1] * carry;
  }
  __syncthreads();

  // coalesced store of segment-local y
  float* wy = ws + WS_Y + (size_t)r * L_LEN + segbase;
  for (int i = 0; i < 16; ++i) {
    wy[i * 256 + tid] = ls_y[i * 256 + tid];
  }
  if (tid == NCHUNK - 1) {
    ws[WS_SEGEND + (size_t)r * NSEG + g] = s;   // segment-end (carry-in 0)
  }
}

// ---------------------------------------------------------------------------
// Serial per-row scan over segment summaries (32 rows x 32 segments).
__global__ void k_carry(float* __restrict__ ws) {
  int r = threadIdx.x;
  if (r >= N_BATCH) return;
  float a4096 = ws[WS_PAR + (size_t)r * P_STRIDE + 3];
  float s = 0.0f;
  for (int g2 = 0; g2 < NSEG; ++g2) {
    ws[WS_CARRY + (size_t)r * NSEG + g2] = s;   // carry INTO segment g2
    s = ws[WS_SEGEND + (size_t)r * NSEG + g2] + a4096 * s;
  }
}

// ---------------------------------------------------------------------------
// Apply carries + truncation fix, gain computer, multiply channels.
__global__ __launch_bounds__(256) void k_finalize(const float* __restrict__ x,
                                                  float* __restrict__ out,
                                                  const float* __restrict__ ws) {
  __shared__ float p16[NCHUNK];     // a^(16j)
  __shared__ float p1[16];          // a^i
  const int g = blockIdx.x;
  const int r = blockIdx.y;
  const int tid = threadIdx.x;

  const float* p = ws + WS_PAR + (size_t)r * P_STRIDE;
  const float alpha = p[0];
  const float aK = p[4];
  const float la = p[5];
  const float T = p[6];
  const float cc = p[7];
  const float W = p[8];
  const float inv4W = 0.25f / W;

  p16[tid] = expf(16.0f * (float)tid * la);
  if (tid < 16) p1[tid] = expf((float)tid * la);
  __syncthreads();

  const size_t segbase = (size_t)g * SEG;
  const float* x0 = x + (size_t)r * 2 * L_LEN;
  const float* x1 = x0 + L_LEN;
  float* o0 = out + (size_t)r * 2 * L_LEN;
  float* o1 = o0 + L_LEN;
  const float* wy = ws + WS_Y + (size_t)r * L_LEN;
  const float carry = ws[WS_CARRY + (size_t)r * NSEG + g];

  for (int i = 0; i < 16; ++i) {
    int off = i * 256 + tid;
    size_t t = segbase + (size_t)off;
    if (i + 1 < 16) {               // software prefetch next strip
      __builtin_prefetch(&x0[t + 256], 0, 0);
      __builtin_prefetch(&x1[t + 256], 0, 0);
    }
    // global y[t] = y_local[t] + a^(off+1) * carry_in(segment)
    float apo = alpha * p16[off >> 4] * p1[off & 15];
    float y = wy[t] + apo * carry;
    float env = y;
    if (t >= (size_t)K_IIR) {       // envelope[t] = y[t] - a^K * y[t-K]
      size_t tk = t - (size_t)K_IIR;
      int off2 = (int)(tk & (SEG - 1));
      int g2 = (int)(tk >> 12);     // SEG == 2^12
      float c2 = ws[WS_CARRY + (size_t)r * NSEG + g2];
      float yt = wy[tk] + alpha * p16[off2 >> 4] * p1[off2 & 15] * c2;
      env = y - aK * yt;
    }
    // gain computer (quadratic knee)
    float le = logf(env + EPSV);
    float d = le - T;
    float lg;
    if (le >= T + W) {
      lg = cc * d;
    } else if (le < T - W) {
      lg = 0.0f;
    } else {
      float q = d + W;
      lg = cc * q * q * inv4W;
    }
    float gain = expf(lg);
    o0[t] = gain * x0[t];
    o1[t] = gain * x1[t];
  }
}

// ---------------------------------------------------------------------------
extern "C" void kernel_launch(void* const* d_in, const int* in_sizes, int n_in,
                              void* d_out, int out_size, void* d_ws, size_t ws_size,
                              hipStream_t stream) {
  const float* x  = (const float*)d_in[0];  // (32, 2, 131072)
  const float* za = (const float*)d_in[1];  // (32, 1)
  const float* lt = (const float*)d_in[2];  // (32, 1)
  const float* lr = (const float*)d_in[3];  // (32, 1)
  const float* lk = (const float*)d_in[4];  // (32, 1)
  float* out = (float*)d_out;               // (32, 2, 131072) f32
  float* ws = (float*)d_ws;                 // needs ~16.8 MB

  k_params<<<1, 32, 0, stream>>>(za, lt, lr, lk, ws);
  k_localscan<<<dim3(NSEG, N_BATCH), 256, 0, stream>>>(x, ws);
  k_carry<<<1, 32, 0, stream>>>(ws);
  k_finalize<<<dim3(NSEG, N_BATCH), 256, 0, stream>>>(x, out, ws);
}